// mLSTMCell_34986803593795
// MI455X (gfx1250) — compile-verified
//
#include <hip/hip_runtime.h>
#include <math.h>

#define SLEN 2048
#define EDIM 1024
#define NH 8
#define DHEAD 128
#define EPSV 1e-6f

typedef float v2f __attribute__((ext_vector_type(2)));
typedef float v8f __attribute__((ext_vector_type(8)));

// ---------------------------------------------------------------------------
// Kernel 1 (WMMA): gate preactivations as a (2048 x 3072) @ (3072 x 16) GEMM.
// One wave per 16 rows. B columns 0-7 = Wi, 8-15 = Wf (fused i/f gates).
// Epilogue adds bias and applies log_sigmoid for the f-gate.
//   ipre[h*S+s] = [q|k|v](s) . Wi[:,h] + bi[h]
//   logfg[h*S+s] = log_sigmoid([q|k|v](s) . Wf[:,h] + bf[h])
// ---------------------------------------------------------------------------
__global__ __launch_bounds__(32)
void gates_wmma_kernel(const float* __restrict__ q, const float* __restrict__ k,
                       const float* __restrict__ v, const float* __restrict__ Wi,
                       const float* __restrict__ bi, const float* __restrict__ Wf,
                       const float* __restrict__ bf, float* __restrict__ ipre,
                       float* __restrict__ logfg) {
  const int s0 = blockIdx.x * 16;
  const int lane = threadIdx.x;
  const int li = lane & 15;
  const int hl = lane >> 4;

  v8f acc;
  #pragma unroll
  for (int j = 0; j < 8; ++j) acc[j] = 0.f;

  // B-fragment source: column n = li selects Wi (n<8) or Wf (n-8).
  // W is (3E, 8) row-major: element (e, col) at W[e*8 + col].
  const float* wptr = ((li < 8) ? (Wi + li) : (Wf + (li - 8))) + hl * 16;

  const float* segs[3] = {q, k, v};
  #pragma unroll
  for (int seg = 0; seg < 3; ++seg) {
    const float* arow = segs[seg] + (size_t)(s0 + li) * EDIM + hl * 2;
    #pragma unroll 16
    for (int kc = 0; kc < 256; ++kc) {
      v2f a = *(const v2f*)(arow + kc * 4);    // gate_in[s0+li][kd0 .. kd0+1]
      v2f b;
      b.x = wptr[0];                            // W[kd0][col]
      b.y = wptr[8];                            // W[kd0+1][col]
      wptr += 32;
      acc = __builtin_amdgcn_wmma_f32_16x16x4_f32(false, a, false, b,
                                                  (short)0, acc, false, false);
    }
  }

  // acc[j] = dot for row r = s0 + j + 8*hl, output column n = li
  #pragma unroll
  for (int j = 0; j < 8; ++j) {
    const int r = s0 + j + 8 * hl;
    float x = acc[j];
    if (li < 8) {
      ipre[li * SLEN + r] = x + bi[li];
    } else {
      float f = x + bf[li - 8];
      // log_sigmoid(x) = min(x,0) - log1p(exp(-|x|))
      logfg[(li - 8) * SLEN + r] = fminf(f, 0.f) - log1pf(__expf(-fabsf(f)));
    }
  }
}

// ---------------------------------------------------------------------------
// Kernel 2: per-head inclusive cumsum of logfg -> cum;  g = ipre - cum
// one wave per head (block = 8 waves)
// ---------------------------------------------------------------------------
__global__ void scan_kernel(const float* __restrict__ ipre, const float* __restrict__ logfg,
                            float* __restrict__ cum, float* __restrict__ g) {
  int h = threadIdx.x >> 5;
  int lane = threadIdx.x & 31;
  float carry = 0.f;
  for (int base = 0; base < SLEN; base += 32) {
    float x = logfg[h * SLEN + base + lane];
    #pragma unroll
    for (int off = 1; off < 32; off <<= 1) {
      float y = __shfl_up(x, off);
      if (lane >= off) x += y;
    }
    x += carry;
    cum[h * SLEN + base + lane] = x;
    g[h * SLEN + base + lane] = ipre[h * SLEN + base + lane] - x;
    carry = __shfl(x, 31);
  }
}

// ---------------------------------------------------------------------------
// Kernel 3: flash-style mLSTM attention, one wave per (head, 16-row tile).
// Score tile computed transposed (S^T = K * Q^T) so each lane owns one row.
// fp32 WMMA 16x16x4 for both GEMMs; next-tile K/V prefetched.
// ---------------------------------------------------------------------------
__global__ __launch_bounds__(32)
void mlstm_attn_kernel(const float* __restrict__ q, const float* __restrict__ k,
                       const float* __restrict__ v, const float* __restrict__ cumv,
                       const float* __restrict__ g, const float* __restrict__ lnscale,
                       float* __restrict__ out) {
  const int rt = blockIdx.x;        // row tile (0..127)
  const int h  = blockIdx.y;        // head
  const int lane = threadIdx.x;
  const int li = lane & 15;
  const int hl = lane >> 4;         // half-wave index
  const int r0 = rt * 16;
  const int iglob = r0 + li;        // this lane's output row (replicated per half)
  const float scaleq = 0.08838834764831845f;  // 1/sqrt(128)

  // Preload Q fragments (B operand of gemm1): bq[kc] = {Q[r0+li][kc*4+2*hl], +1} * scaleq
  v2f bq[32];
  {
    const float* qrow = q + (size_t)(r0 + li) * EDIM + h * DHEAD + hl * 2;
    #pragma unroll
    for (int kc = 0; kc < 32; ++kc) {
      v2f t = *(const v2f*)(qrow + kc * 4);
      bq[kc].x = t.x * scaleq;
      bq[kc].y = t.y * scaleq;
    }
  }

  // O accumulators: O[dchunk][j] holds out row (j + 8*hl), col d = dchunk*16 + li
  float O[8][8];
  #pragma unroll
  for (int c = 0; c < 8; ++c)
    #pragma unroll
    for (int j = 0; j < 8; ++j) O[c][j] = 0.f;

  float mg = -INFINITY;   // running max of g[j], j<=iglob   (per-lane row stat)
  float ssum = 0.f;       // running signed sum of C row

  const float* gh = g + h * SLEN;

  for (int ct = 0; ct <= rt; ++ct) {
    const int c0 = ct * 16;

    // prefetch next column tile's K/V rows (lane li -> row c0+16+li)
    if (ct < rt) {
      const float* nk = k + (size_t)(c0 + 16 + li) * EDIM + h * DHEAD;
      const float* nv = v + (size_t)(c0 + 16 + li) * EDIM + h * DHEAD;
      __builtin_prefetch(nk, 0, 3);
      __builtin_prefetch(nk + 64, 0, 3);
      __builtin_prefetch(nv, 0, 3);
      __builtin_prefetch(nv + 64, 0, 3);
    }

    // ---- gemm1: S^T tile = K_tile (A) x Q_tile^T (B), K over DH=128 ----
    v8f acc;
    #pragma unroll
    for (int j = 0; j < 8; ++j) acc[j] = 0.f;
    const float* krow = k + (size_t)(c0 + li) * EDIM + h * DHEAD + hl * 2;
    #pragma unroll
    for (int kc = 0; kc < 32; ++kc) {
      v2f a = *(const v2f*)(krow + kc * 4);
      acc = __builtin_amdgcn_wmma_f32_16x16x4_f32(false, a, false, bq[kc],
                                                  (short)0, acc, false, false);
    }
    // lane holds S[i=li][j = c0 + jj + 8*hl] in acc[jj]

    // column decay terms g[j] and causal mask
    float gcol[8];
    bool  okm[8];
    float mloc = -INFINITY;
    #pragma unroll
    for (int jj = 0; jj < 8; ++jj) {
      int jg = c0 + jj + 8 * hl;
      gcol[jj] = gh[jg];
      okm[jj] = (ct < rt) || (jg <= iglob);
      mloc = fmaxf(mloc, okm[jj] ? gcol[jj] : -INFINITY);
    }
    mloc = fmaxf(mloc, __shfl_xor(mloc, 16));      // combine the two j-halves
    float mgnew = fmaxf(mg, mloc);
    float rescale = __expf(mg - mgnew);            // == 0 on the first tile (mg=-inf)
    mg = mgnew;

    // P = S * exp(g - mg), masked; row sum
    float p[8];
    float psum = 0.f;
    #pragma unroll
    for (int jj = 0; jj < 8; ++jj) {
      float d = okm[jj] ? __expf(gcol[jj] - mgnew) : 0.f;
      p[jj] = acc[jj] * d;
      psum += p[jj];
    }
    psum += __shfl_xor(psum, 16);
    ssum = ssum * rescale + psum;

    // per-row rescale factors for O (row of O vgpr j is j + 8*hl)
    float rs[8];
    #pragma unroll
    for (int j = 0; j < 8; ++j) rs[j] = __shfl(rescale, j + 8 * hl);

    // A fragments for gemm2 from P (C-layout -> A-layout via shuffles).
    // Dest lane wants P[li][cA+comp], cA = kc*4 + 2*hl; holder lane = li + 16*(c>>3), vgpr = c&7.
    v2f ap[4];
    #pragma unroll
    for (int kc = 0; kc < 4; ++kc) {
      const int ca0 = kc * 4;       // half 0 columns ca0, ca0+1
      const int ca1 = kc * 4 + 2;   // half 1 columns ca1, ca1+1
      float x0 = __shfl(p[ca0 & 7],       li + 16 * (ca0 >> 3));
      float y0 = __shfl(p[(ca0 + 1) & 7], li + 16 * ((ca0 + 1) >> 3));
      float x1 = __shfl(p[ca1 & 7],       li + 16 * (ca1 >> 3));
      float y1 = __shfl(p[(ca1 + 1) & 7], li + 16 * ((ca1 + 1) >> 3));
      ap[kc].x = hl ? x1 : x0;
      ap[kc].y = hl ? y1 : y0;
    }

    // ---- gemm2: O += P x V_tile (8 column chunks of 16), with per-row rescale ----
    const float* vbase = v + (size_t)c0 * EDIM + h * DHEAD;
    #pragma unroll
    for (int c = 0; c < 8; ++c) {
      v8f co;
      #pragma unroll
      for (int j = 0; j < 8; ++j) co[j] = O[c][j] * rs[j];
      #pragma unroll
      for (int kc = 0; kc < 4; ++kc) {
        const int jr = kc * 4 + hl * 2;
        v2f bv;
        bv.x = vbase[(size_t)jr * EDIM + c * 16 + li];
        bv.y = vbase[(size_t)(jr + 1) * EDIM + c * 16 + li];
        co = __builtin_amdgcn_wmma_f32_16x16x4_f32(false, ap[kc], false, bv,
                                                   (short)0, co, false, false);
      }
      #pragma unroll
      for (int j = 0; j < 8; ++j) O[c][j] = co[j];
    }
  }

  // normalizer: max(|ssum|, exp(-cum_i - mg_i)) + EPS   (per-lane row li)
  float ci = cumv[h * SLEN + iglob];
  float nrm = fmaxf(fabsf(ssum), __expf(-ci - mg)) + EPSV;
  float inv = 1.f / nrm;

  // per-head LayerNorm over DH + store
  const float* lns = lnscale + h * DHEAD;
  float* orow = out + (size_t)r0 * EDIM + h * DHEAD;
  #pragma unroll
  for (int j = 0; j < 8; ++j) {
    float invj = __shfl(inv, j + 8 * hl);   // normalizer for row j + 8*hl
    float vals[8];
    float sum = 0.f, sq = 0.f;
    #pragma unroll
    for (int c = 0; c < 8; ++c) {
      float val = O[c][j] * invj;
      vals[c] = val;
      sum += val;
      sq += val * val;
    }
    #pragma unroll
    for (int off = 1; off < 16; off <<= 1) {   // butterfly within the 16-lane half
      sum += __shfl_xor(sum, off);
      sq  += __shfl_xor(sq, off);
    }
    float mu = sum * (1.f / 128.f);
    float var = sq * (1.f / 128.f) - mu * mu;
    float rstd = rsqrtf(var + EPSV);
    const int row = j + 8 * hl;
    #pragma unroll
    for (int c = 0; c < 8; ++c) {
      float val = (vals[c] - mu) * rstd * lns[c * 16 + li];
      orow[(size_t)row * EDIM + c * 16 + li] = val;
    }
  }
}

// ---------------------------------------------------------------------------
extern "C" void kernel_launch(void* const* d_in, const int* in_sizes, int n_in,
                              void* d_out, int out_size, void* d_ws, size_t ws_size,
                              hipStream_t stream) {
  const float* q  = (const float*)d_in[0];
  const float* k  = (const float*)d_in[1];
  const float* v  = (const float*)d_in[2];
  const float* Wi = (const float*)d_in[3];
  const float* bi = (const float*)d_in[4];
  const float* Wf = (const float*)d_in[5];
  const float* bf = (const float*)d_in[6];
  const float* ln = (const float*)d_in[7];
  float* out = (float*)d_out;

  float* ws    = (float*)d_ws;
  float* ipre  = ws;                    // NH*S
  float* logfg = ws + 1 * NH * SLEN;    // NH*S
  float* cumb  = ws + 2 * NH * SLEN;    // NH*S
  float* gb    = ws + 3 * NH * SLEN;    // NH*S   (total 256 KB)

  gates_wmma_kernel<<<dim3(SLEN / 16), 32, 0, stream>>>(
      q, k, v, Wi, bi, Wf, bf, ipre, logfg);
  scan_kernel<<<1, 256, 0, stream>>>(ipre, logfg, cumb, gb);

  dim3 grid(SLEN / 16, NH);
  mlstm_attn_kernel<<<grid, 32, 0, stream>>>(q, k, v, cumb, gb, ln, out);
}